// GAT_6536940224658
// MI455X (gfx1250) — compile-verified
//
#include <hip/hip_runtime.h>
#include <math.h>

// ---------------- problem constants (match reference) ----------------
#define NNODES   50000
#define NEDGES   1600000
#define ETOT     (NEDGES + NNODES)   // edges + self loops
#define IN_DIM   128
#define HID      64
#define HEADS    8
#define HD       8
#define RH       32
#define NEG_SLOPE 0.2f
#define LN_EPS   1e-5f
#define CBLK     196                 // ceil(NNODES/256)

typedef float v2f __attribute__((ext_vector_type(2)));
typedef float v8f __attribute__((ext_vector_type(8)));

// =====================================================================
// WMMA f32 GEMM with LDS staging:
//   out[M x NC] = A[M x K] * W[K x NC]   (all row-major)
// grid.x = M/16 row tiles; blockDim.x = 32*(NC/16)  (one wave per col tile)
// A tile (16 x K) and full W panel (K x NC) staged in LDS with coalesced
// float4 loads; WMMA fragments then come from ds_load.
// V_WMMA_F32_16X16X4_F32 layouts per CDNA5 ISA 7.12.2:
//   A 16x4 f32 : lanes 0-15 rows M, VGPR{0,1}=K{0,1}; lanes16-31 K{2,3}
//   B 4x16 f32 : row striped across lanes; lane half selects K pair
//   C/D 16x16  : VGPR v -> row v + 8*(lane>>4), col = lane&15
// =====================================================================
template<int K, int NC>
__global__ void gat_gemm_wmma(const float* __restrict__ A,
                              const float* __restrict__ W,
                              float* __restrict__ out) {
    // +2 pad keeps float2 alignment and breaks the 64-bank conflict
    __shared__ float As[16][K + 2];
    __shared__ float Ws[K][NC];

    constexpr int NT = 32 * (NC / 16);
    const int tid  = threadIdx.x;
    const int row0 = blockIdx.x << 4;

    // ---- stage A tile (16 x K), coalesced b128 ----
    constexpr int AV4 = 16 * K / 4;
    for (int i = tid; i < AV4; i += NT) {
        int r = (i * 4) / K, c = (i * 4) % K;
        float4 v = *(const float4*)(A + (size_t)(row0 + r) * K + c);
        As[r][c] = v.x; As[r][c + 1] = v.y; As[r][c + 2] = v.z; As[r][c + 3] = v.w;
    }
    // ---- stage W panel (K x NC), coalesced b128 ----
    constexpr int WV4 = K * NC / 4;
    for (int i = tid; i < WV4; i += NT) {
        int r = (i * 4) / NC, c = (i * 4) % NC;
        float4 v = *(const float4*)(W + (size_t)r * NC + c);
        Ws[r][c] = v.x; Ws[r][c + 1] = v.y; Ws[r][c + 2] = v.z; Ws[r][c + 3] = v.w;
    }
    __syncthreads();

    const int lane  = tid & 31;
    const int wave  = tid >> 5;
    const int col0  = wave << 4;
    const int m     = lane & 15;
    const int khalf = (lane >> 4) << 1;       // 0 or 2

    v8f acc = {0.f, 0.f, 0.f, 0.f, 0.f, 0.f, 0.f, 0.f};
#pragma unroll
    for (int k = 0; k < K; k += 4) {
        v2f a, b;
        a.x = As[m][k + khalf];
        a.y = As[m][k + khalf + 1];
        b.x = Ws[k + khalf][col0 + m];
        b.y = Ws[k + khalf + 1][col0 + m];
        // (neg_a, A, neg_b, B, c_mod, C, reuse_a, reuse_b)
        acc = __builtin_amdgcn_wmma_f32_16x16x4_f32(false, a, false, b,
                                                    (short)0, acc, false, false);
    }

    const int colw  = col0 + (lane & 15);
    const int rbase = row0 + ((lane >> 4) << 3);
#pragma unroll
    for (int v = 0; v < 8; ++v)
        out[(size_t)(rbase + v) * NC + colw] = acc[v];
}

// ---------------- per-(node,head) attention scores -------------------
__global__ void gat_scores(const float* __restrict__ xw,
                           const float* __restrict__ a_src,
                           const float* __restrict__ a_dst,
                           float* __restrict__ s_src,
                           float* __restrict__ s_dst) {
    int idx = blockIdx.x * blockDim.x + threadIdx.x;     // n*8 + h
    if (idx >= NNODES * HEADS) return;
    int h = idx & 7, n = idx >> 3;
    const float* v = xw + (size_t)n * HID + h * HD;
    float ss = 0.f, sd = 0.f;
#pragma unroll
    for (int d = 0; d < HD; ++d) {
        ss = fmaf(v[d], a_src[h * HD + d], ss);
        sd = fmaf(v[d], a_dst[h * HD + d], sd);
    }
    s_src[idx] = ss;
    s_dst[idx] = sd;
}

// ---------------- init per-layer accumulators ------------------------
__global__ void gat_init(float* __restrict__ agg,
                         float* __restrict__ mbuf,
                         float* __restrict__ dbuf) {
    int idx = blockIdx.x * blockDim.x + threadIdx.x;
    if (idx < NNODES * HID) agg[idx] = 0.f;
    if (idx < NNODES * HEADS) {
        ((unsigned int*)mbuf)[idx] = 0xFF800000u;   // -inf
        dbuf[idx] = 0.f;
    }
}

// ---------------- helpers --------------------------------------------
__device__ __forceinline__ void atomicMaxF(float* addr, float val) {
    if (val >= 0.f) atomicMax((int*)addr, __float_as_int(val));
    else            atomicMin((unsigned int*)addr, __float_as_uint(val));
}
__device__ __forceinline__ float lrelu(float x) {
    return x > 0.f ? x : NEG_SLOPE * x;
}

// ---------------- edge pass 1: segment max (1 thread / edge) ---------
__global__ void gat_edge_max(const int* __restrict__ esrc,
                             const int* __restrict__ edst,
                             const float* __restrict__ s_src,
                             const float* __restrict__ s_dst,
                             float* __restrict__ mbuf) {
    int e = blockIdx.x * blockDim.x + threadIdx.x;
    if (e >= ETOT) return;
    int s = (e < NEDGES) ? esrc[e] : (e - NEDGES);
    int d = (e < NEDGES) ? edst[e] : (e - NEDGES);
    float4 a0 = *(const float4*)(s_src + (size_t)s * HEADS);
    float4 a1 = *(const float4*)(s_src + (size_t)s * HEADS + 4);
    float4 b0 = *(const float4*)(s_dst + (size_t)d * HEADS);
    float4 b1 = *(const float4*)(s_dst + (size_t)d * HEADS + 4);
    float* mrow = mbuf + (size_t)d * HEADS;
    atomicMaxF(mrow + 0, lrelu(a0.x + b0.x));
    atomicMaxF(mrow + 1, lrelu(a0.y + b0.y));
    atomicMaxF(mrow + 2, lrelu(a0.z + b0.z));
    atomicMaxF(mrow + 3, lrelu(a0.w + b0.w));
    atomicMaxF(mrow + 4, lrelu(a1.x + b1.x));
    atomicMaxF(mrow + 5, lrelu(a1.y + b1.y));
    atomicMaxF(mrow + 6, lrelu(a1.z + b1.z));
    atomicMaxF(mrow + 7, lrelu(a1.w + b1.w));
}

// ---------------- edge pass 2: softmax denominator (1 thread/edge) ---
__global__ void gat_edge_den(const int* __restrict__ esrc,
                             const int* __restrict__ edst,
                             const float* __restrict__ s_src,
                             const float* __restrict__ s_dst,
                             const float* __restrict__ mbuf,
                             float* __restrict__ dbuf) {
    int e = blockIdx.x * blockDim.x + threadIdx.x;
    if (e >= ETOT) return;
    int s = (e < NEDGES) ? esrc[e] : (e - NEDGES);
    int d = (e < NEDGES) ? edst[e] : (e - NEDGES);
    const float* ar = s_src + (size_t)s * HEADS;
    const float* br = s_dst + (size_t)d * HEADS;
    const float* mr = mbuf + (size_t)d * HEADS;
    float* dr = dbuf + (size_t)d * HEADS;
#pragma unroll
    for (int h = 0; h < HEADS; ++h) {
        float mm = mr[h];
        if (!(mm > -1e38f)) mm = 0.f;
        atomicAdd(dr + h, expf(lrelu(ar[h] + br[h]) - mm));
    }
}

// ---------------- edge pass 3: weighted aggregation ------------------
// 64 threads per edge (2 waves); indices + per-head scalars loaded once
// and wave-broadcast via __shfl (wave32).
__global__ void gat_edge_agg(const int* __restrict__ esrc,
                             const int* __restrict__ edst,
                             const float* __restrict__ s_src,
                             const float* __restrict__ s_dst,
                             const float* __restrict__ mbuf,
                             const float* __restrict__ dbuf,
                             const float* __restrict__ xw,
                             float* __restrict__ agg) {
    int idx = blockIdx.x * blockDim.x + threadIdx.x;     // e*64 + c
    int e = idx >> 6;
    if (e >= ETOT) return;                               // wave-uniform
    int lane = threadIdx.x & 31;
    int c = idx & 63;
    int h = c >> 3;

    int s = 0, d = 0;
    if (lane == 0) {
        s = (e < NEDGES) ? esrc[e] : (e - NEDGES);
        d = (e < NEDGES) ? edst[e] : (e - NEDGES);
    }
    s = __shfl(s, 0, 32);
    d = __shfl(d, 0, 32);

    float vs = 0.f, vd = 0.f, mm = 0.f, den = 0.f;
    if ((lane & 7) == 0) {                               // one leader per head group
        vs  = s_src[(size_t)s * HEADS + h];
        vd  = s_dst[(size_t)d * HEADS + h];
        mm  = mbuf[(size_t)d * HEADS + h];
        den = dbuf[(size_t)d * HEADS + h];
    }
    int ldr = lane & 24;
    vs  = __shfl(vs,  ldr, 32);
    vd  = __shfl(vd,  ldr, 32);
    mm  = __shfl(mm,  ldr, 32);
    den = __shfl(den, ldr, 32);

    float v = lrelu(vs + vd);
    if (!(mm > -1e38f)) mm = 0.f;
    float alpha = expf(v - mm) / fmaxf(den, 1e-16f);
    atomicAdd(&agg[(size_t)d * HID + c], alpha * xw[(size_t)s * HID + c]);
}

// ---------------- bias + layernorm + relu + residual (wave/node) -----
__global__ void gat_node_post(const float* __restrict__ agg,
                              const float* __restrict__ bias,
                              const float* __restrict__ g,
                              const float* __restrict__ b,
                              float* __restrict__ h, int layer) {
    int node = (blockIdx.x * blockDim.x + threadIdx.x) >> 5;
    int lane = threadIdx.x & 31;
    if (node >= NNODES) return;
    size_t base = (size_t)node * HID;
    float v0 = agg[base + lane]      + bias[lane];
    float v1 = agg[base + lane + 32] + bias[lane + 32];
    float s = v0 + v1;
#pragma unroll
    for (int o = 16; o > 0; o >>= 1) s += __shfl_xor(s, o, 32);
    float mu = s * (1.f / 64.f);
    float d0 = v0 - mu, d1 = v1 - mu;
    float vs = d0 * d0 + d1 * d1;
#pragma unroll
    for (int o = 16; o > 0; o >>= 1) vs += __shfl_xor(vs, o, 32);
    float inv = rsqrtf(vs * (1.f / 64.f) + LN_EPS);
    float r0 = fmaxf(fmaf(d0 * inv, g[lane],      b[lane]),      0.f);
    float r1 = fmaxf(fmaf(d1 * inv, g[lane + 32], b[lane + 32]), 0.f);
    if (layer) { r0 += h[base + lane]; r1 += h[base + lane + 32]; }
    h[base + lane]      = r0;
    h[base + lane + 32] = r1;
}

// ---------------- row L2-normalize (wave/node, in place) -------------
__global__ void gat_normalize(float* __restrict__ h) {
    int node = (blockIdx.x * blockDim.x + threadIdx.x) >> 5;
    int lane = threadIdx.x & 31;
    if (node >= NNODES) return;
    size_t base = (size_t)node * HID;
    float v0 = h[base + lane], v1 = h[base + lane + 32];
    float n2 = v0 * v0 + v1 * v1;
#pragma unroll
    for (int o = 16; o > 0; o >>= 1) n2 += __shfl_xor(n2, o, 32);
    float inv = 1.f / fmaxf(sqrtf(n2), 1e-12f);
    h[base + lane]      = v0 * inv;
    h[base + lane + 32] = v1 * inv;
}

// ---------------- angle head: +b1, LN(64), relu, dot w2 --------------
__global__ void gat_angle_post(const float* __restrict__ a1,
                               const float* __restrict__ b1,
                               const float* __restrict__ g,
                               const float* __restrict__ b,
                               const float* __restrict__ w2,
                               const float* __restrict__ b2,
                               float* __restrict__ angles) {
    int node = (blockIdx.x * blockDim.x + threadIdx.x) >> 5;
    int lane = threadIdx.x & 31;
    if (node >= NNODES) return;
    size_t base = (size_t)node * HID;
    float v0 = a1[base + lane]      + b1[lane];
    float v1 = a1[base + lane + 32] + b1[lane + 32];
    float s = v0 + v1;
#pragma unroll
    for (int o = 16; o > 0; o >>= 1) s += __shfl_xor(s, o, 32);
    float mu = s * (1.f / 64.f);
    float d0 = v0 - mu, d1 = v1 - mu;
    float vs = d0 * d0 + d1 * d1;
#pragma unroll
    for (int o = 16; o > 0; o >>= 1) vs += __shfl_xor(vs, o, 32);
    float inv = rsqrtf(vs * (1.f / 64.f) + LN_EPS);
    float r0 = fmaxf(fmaf(d0 * inv, g[lane],      b[lane]),      0.f);
    float r1 = fmaxf(fmaf(d1 * inv, g[lane + 32], b[lane + 32]), 0.f);
    float dot = fmaf(r0, w2[lane], r1 * w2[lane + 32]);
#pragma unroll
    for (int o = 16; o > 0; o >>= 1) dot += __shfl_xor(dot, o, 32);
    if (lane == 0) angles[node] = dot + b2[0];
}

// ---------------- radius head: +b1, LN(32), relu, dot w2, sigmoid ----
__global__ void gat_rad_post(const float* __restrict__ r1,
                             const float* __restrict__ b1,
                             const float* __restrict__ g,
                             const float* __restrict__ b,
                             const float* __restrict__ w2,
                             const float* __restrict__ b2,
                             float* __restrict__ rr) {
    int node = (blockIdx.x * blockDim.x + threadIdx.x) >> 5;
    int lane = threadIdx.x & 31;
    if (node >= NNODES) return;
    float v = r1[(size_t)node * RH + lane] + b1[lane];
    float s = v;
#pragma unroll
    for (int o = 16; o > 0; o >>= 1) s += __shfl_xor(s, o, 32);
    float mu = s * (1.f / 32.f);
    float d = v - mu;
    float vs = d * d;
#pragma unroll
    for (int o = 16; o > 0; o >>= 1) vs += __shfl_xor(vs, o, 32);
    float inv = rsqrtf(vs * (1.f / 32.f) + LN_EPS);
    float r = fmaxf(fmaf(d * inv, g[lane], b[lane]), 0.f);
    float dot = r * w2[lane];
#pragma unroll
    for (int o = 16; o > 0; o >>= 1) dot += __shfl_xor(dot, o, 32);
    if (lane == 0) {
        float sig = 1.f / (1.f + expf(-(dot + b2[0])));
        rr[node] = 0.9f + 0.2f * sig;
    }
}

// ---------------- coords + deterministic block partial sums ----------
__global__ void gat_coords(const float* __restrict__ angles,
                           const float* __restrict__ rr,
                           float* __restrict__ out,
                           float* __restrict__ partial) {
    __shared__ float sx[256], sy[256];
    int tid = threadIdx.x;
    int n = blockIdx.x * 256 + tid;
    float cx = 0.f, cy = 0.f;
    if (n < NNODES) {
        float a = angles[n], r = rr[n];
        cx = r * cosf(a);
        cy = r * sinf(a);
        out[(size_t)2 * n]     = cx;
        out[(size_t)2 * n + 1] = cy;
    }
    sx[tid] = cx; sy[tid] = cy;
    __syncthreads();
#pragma unroll
    for (int o = 128; o > 0; o >>= 1) {
        if (tid < o) { sx[tid] += sx[tid + o]; sy[tid] += sy[tid + o]; }
        __syncthreads();
    }
    if (tid == 0) { partial[blockIdx.x] = sx[0]; partial[CBLK + blockIdx.x] = sy[0]; }
}

__global__ void gat_reduce_partials(const float* __restrict__ partial,
                                    float* __restrict__ sums) {
    __shared__ float sx[256], sy[256];
    int tid = threadIdx.x;
    sx[tid] = (tid < CBLK) ? partial[tid] : 0.f;
    sy[tid] = (tid < CBLK) ? partial[CBLK + tid] : 0.f;
    __syncthreads();
#pragma unroll
    for (int o = 128; o > 0; o >>= 1) {
        if (tid < o) { sx[tid] += sx[tid + o]; sy[tid] += sy[tid + o]; }
        __syncthreads();
    }
    if (tid == 0) { sums[0] = sx[0]; sums[1] = sy[0]; }
}

__global__ void gat_center(float* __restrict__ out, const float* __restrict__ sums) {
    int n = blockIdx.x * blockDim.x + threadIdx.x;
    if (n >= NNODES) return;
    out[(size_t)2 * n]     -= sums[0] * (1.f / (float)NNODES);
    out[(size_t)2 * n + 1] -= sums[1] * (1.f / (float)NNODES);
}

// =====================================================================
extern "C" void kernel_launch(void* const* d_in, const int* in_sizes, int n_in,
                              void* d_out, int out_size, void* d_ws, size_t ws_size,
                              hipStream_t stream) {
    const float* x        = (const float*)d_in[0];
    const int*   ei       = (const int*)  d_in[1];
    const float* Wc[3]    = {(const float*)d_in[2], (const float*)d_in[3], (const float*)d_in[4]};
    const float* att_src  = (const float*)d_in[5];
    const float* att_dst  = (const float*)d_in[6];
    const float* conv_b   = (const float*)d_in[7];
    const float* ln_g     = (const float*)d_in[8];
    const float* ln_b     = (const float*)d_in[9];
    const float* a_w1     = (const float*)d_in[10];
    const float* a_b1     = (const float*)d_in[11];
    const float* a_ln_g   = (const float*)d_in[12];
    const float* a_ln_b   = (const float*)d_in[13];
    const float* a_w2     = (const float*)d_in[14];
    const float* a_b2     = (const float*)d_in[15];
    const float* r_w1     = (const float*)d_in[16];
    const float* r_b1     = (const float*)d_in[17];
    const float* r_ln_g   = (const float*)d_in[18];
    const float* r_ln_b   = (const float*)d_in[19];
    const float* r_w2     = (const float*)d_in[20];
    const float* r_b2     = (const float*)d_in[21];

    const int* esrc = ei;
    const int* edst = ei + NEDGES;

    // workspace carve-up (floats)
    float* ws     = (float*)d_ws;
    float* h      = ws;                            // N*64
    float* xw     = h    + (size_t)NNODES * HID;   // N*64 (also angle hidden)
    float* agg    = xw   + (size_t)NNODES * HID;   // N*64 (also radius hidden N*32)
    float* ssrc   = agg  + (size_t)NNODES * HID;   // N*8
    float* sdst   = ssrc + (size_t)NNODES * HEADS; // N*8
    float* mbuf   = sdst + (size_t)NNODES * HEADS; // N*8
    float* dbuf   = mbuf + (size_t)NNODES * HEADS; // N*8
    float* angles = dbuf + (size_t)NNODES * HEADS; // N
    float* rr     = angles + NNODES;               // N
    float* part   = rr + NNODES;                   // 2*CBLK
    float* sums   = part + 2 * CBLK;               // 2

    const int rowTiles = NNODES / 16;              // 3125, exact
    const int nhBlocks = (NNODES * HEADS + 255) / 256;
    const int nInit    = (NNODES * HID + 255) / 256;
    const int eBlocks  = (ETOT + 255) / 256;
    const int ecBlocks = (ETOT * HID + 255) / 256;
    const int wvBlocks = (NNODES * 32 + 255) / 256; // wave-per-node

    for (int layer = 0; layer < 3; ++layer) {
        if (layer == 0)
            gat_gemm_wmma<IN_DIM, HID><<<rowTiles, 128, 0, stream>>>(x, Wc[0], xw);
        else
            gat_gemm_wmma<HID, HID><<<rowTiles, 128, 0, stream>>>(h, Wc[layer], xw);

        gat_scores<<<nhBlocks, 256, 0, stream>>>(xw, att_src + layer * HEADS * HD,
                                                 att_dst + layer * HEADS * HD, ssrc, sdst);
        gat_init<<<nInit, 256, 0, stream>>>(agg, mbuf, dbuf);
        gat_edge_max<<<eBlocks, 256, 0, stream>>>(esrc, edst, ssrc, sdst, mbuf);
        gat_edge_den<<<eBlocks, 256, 0, stream>>>(esrc, edst, ssrc, sdst, mbuf, dbuf);
        gat_edge_agg<<<ecBlocks, 256, 0, stream>>>(esrc, edst, ssrc, sdst, mbuf, dbuf, xw, agg);
        gat_node_post<<<wvBlocks, 256, 0, stream>>>(agg, conv_b + layer * HID,
                                                    ln_g + layer * HID, ln_b + layer * HID,
                                                    h, layer);
    }

    gat_normalize<<<wvBlocks, 256, 0, stream>>>(h);

    gat_gemm_wmma<HID, HID><<<rowTiles, 128, 0, stream>>>(h, a_w1, xw);   // angle hidden
    gat_gemm_wmma<HID, RH><<<rowTiles, 64, 0, stream>>>(h, r_w1, agg);    // radius hidden

    gat_angle_post<<<wvBlocks, 256, 0, stream>>>(xw, a_b1, a_ln_g, a_ln_b, a_w2, a_b2, angles);
    gat_rad_post<<<wvBlocks, 256, 0, stream>>>(agg, r_b1, r_ln_g, r_ln_b, r_w2, r_b2, rr);

    gat_coords<<<CBLK, 256, 0, stream>>>(angles, rr, (float*)d_out, part);
    gat_reduce_partials<<<1, 256, 0, stream>>>(part, sums);
    gat_center<<<(NNODES + 255) / 256, 256, 0, stream>>>((float*)d_out, sums);
}